// STAN_2224793059992
// MI455X (gfx1250) — compile-verified
//
#include <hip/hip_runtime.h>
#include <math.h>

#define NUM_NODES_C 50000
#define B_C 4
#define T_C 16
#define NF_C 4
#define NTOT_C (B_C * NUM_NODES_C)   /* 200000 */
#define HOR_C 8

typedef __attribute__((ext_vector_type(2))) float v2f;
typedef __attribute__((ext_vector_type(8))) float v8f;

__device__ __forceinline__ float elu1(float x) { return x > 0.f ? x : (expf(x) - 1.f); }
__device__ __forceinline__ float sigm(float x) { return 1.f / (1.f + expf(-x)); }

// ---------------------------------------------------------------------------
// K1: z1[n][0..31] = Xf[n] @ fc1_W.T + b   (K = T*NF = 64), and zero agg1.
// One wave = 16 nodes (one WMMA M-tile). f32 WMMA 16x16x4, 16 K-steps, 2 N-tiles.
// A layout (32-bit, 16x4): lane l holds M = l&15, K = 4*t + 2*(l>>4) + vgpr.
// ---------------------------------------------------------------------------
__global__ __launch_bounds__(128) void k_gemm1(const float* __restrict__ X,
    const float* __restrict__ W, const float* __restrict__ bias,
    float* __restrict__ z, float* __restrict__ agg)
{
    const int lane = threadIdx.x & 31;
    const int tile = blockIdx.x * 4 + (threadIdx.x >> 5);   // 0..12499
    const int base = tile * 16;
    const int mrow = lane & 15;
    const int kb   = (lane >> 4) * 2;                       // 0 or 2
    const int nf   = base + mrow;
    const int bb   = nf / NUM_NODES_C;                      // tile never crosses batch
    const int nn   = nf - bb * NUM_NODES_C;

    // A operand: Xf[node][k] = X[b][t][n][f], k = t*4+f; per k-step load float2.
    v2f a[T_C];
    const float* xp = X + ((size_t)bb * T_C * NUM_NODES_C + nn) * NF_C + kb;
#pragma unroll
    for (int t = 0; t < T_C; ++t) {
        const float* p = xp + (size_t)t * (NUM_NODES_C * NF_C);
        a[t].x = p[0]; a[t].y = p[1];
    }
    const int mo = (lane >> 4) * 8;
#pragma unroll
    for (int nt = 0; nt < 2; ++nt) {
        const int col = nt * 16 + mrow;                     // output channel (N = l&15)
        const float* wr = W + (size_t)col * (T_C * NF_C) + kb; // B[k][n] = W[n][k]
        v8f acc = {};
#pragma unroll
        for (int t = 0; t < T_C; ++t) {
            v2f bv; bv.x = wr[t * 4]; bv.y = wr[t * 4 + 1];
            acc = __builtin_amdgcn_wmma_f32_16x16x4_f32(false, a[t], false, bv,
                                                        (short)0, acc, false, false);
        }
        const float bc = bias[col];
#pragma unroll
        for (int v = 0; v < 8; ++v)                         // D: M = v + 8*(l>>4), N = col
            z[(size_t)(base + mo + v) * 32 + col] = acc[v] + bc;
    }
    // zero this tile's 16 agg rows (512 floats = 128 float4 = 32 lanes * 4)
    float4 z4 = make_float4(0.f, 0.f, 0.f, 0.f);
    float4* ap = (float4*)(agg + (size_t)base * 32);
#pragma unroll
    for (int i = 0; i < 4; ++i) ap[lane + 32 * i] = z4;
}

// ---------------------------------------------------------------------------
// Edge pass: one wave per edge, one lane per channel. Edge endpoints are
// wave-uniform -> readfirstlane so clang emits SMEM scalar loads (KMcnt path),
// keeping LOADcnt/STOREcnt free for the row gathers + atomics. Gather two
// 128B z-rows (L2-resident), 64-wide attention dot via shfl-xor reduction,
// leaky-relu, then 32 no-return f32 atomics into agg[dst].
// ---------------------------------------------------------------------------
__global__ __launch_bounds__(256) void k_edge(const int* __restrict__ src,
    const int* __restrict__ dst, const float* __restrict__ z,
    const float* __restrict__ aW, const float* __restrict__ ab,
    float* __restrict__ agg, int nE)
{
    const int e = __builtin_amdgcn_readfirstlane(blockIdx.x * 8 + (threadIdx.x >> 5));
    if (e >= nE) return;
    const int lane = threadIdx.x & 31;
    const int s = src[e];                                   // scalar (s_load) loads
    const int d = dst[e];
    const float zs = z[(size_t)s * 32 + lane];
    const float zd = z[(size_t)d * 32 + lane];
    float p = zs * aW[lane] + zd * aW[32 + lane];
#pragma unroll
    for (int off = 16; off; off >>= 1) p += __shfl_xor(p, off, 32);
    float ev = p + ab[0];
    ev = ev > 0.f ? ev : 0.01f * ev;                        // leaky_relu(0.01)
    unsafeAtomicAdd(agg + (size_t)d * 32 + lane, zs * ev);  // global_atomic_add_f32
}

// ---------------------------------------------------------------------------
// K3: z2 = elu(agg1) @ fc2_W.T + b  (K = 32), and zero agg2.
// ---------------------------------------------------------------------------
__global__ __launch_bounds__(128) void k_gemm2(const float* __restrict__ agg1,
    const float* __restrict__ W, const float* __restrict__ bias,
    float* __restrict__ z, float* __restrict__ agg2)
{
    const int lane = threadIdx.x & 31;
    const int tile = blockIdx.x * 4 + (threadIdx.x >> 5);
    const int base = tile * 16;
    const int mrow = lane & 15;
    const int kb   = (lane >> 4) * 2;

    v2f a[8];
    const float* hp = agg1 + (size_t)(base + mrow) * 32 + kb;
#pragma unroll
    for (int ks = 0; ks < 8; ++ks) {
        a[ks].x = elu1(hp[ks * 4]);
        a[ks].y = elu1(hp[ks * 4 + 1]);
    }
    const int mo = (lane >> 4) * 8;
#pragma unroll
    for (int nt = 0; nt < 2; ++nt) {
        const int col = nt * 16 + mrow;
        const float* wr = W + (size_t)col * 32 + kb;
        v8f acc = {};
#pragma unroll
        for (int ks = 0; ks < 8; ++ks) {
            v2f bv; bv.x = wr[ks * 4]; bv.y = wr[ks * 4 + 1];
            acc = __builtin_amdgcn_wmma_f32_16x16x4_f32(false, a[ks], false, bv,
                                                        (short)0, acc, false, false);
        }
        const float bc = bias[col];
#pragma unroll
        for (int v = 0; v < 8; ++v)
            z[(size_t)(base + mo + v) * 32 + col] = acc[v] + bc;
    }
    float4 z4 = make_float4(0.f, 0.f, 0.f, 0.f);
    float4* ap = (float4*)(agg2 + (size_t)base * 32);
#pragma unroll
    for (int i = 0; i < 4; ++i) ap[lane + 32 * i] = z4;
}

// ---------------------------------------------------------------------------
// K5: head. h2 = elu(agg2); gi = h2 @ Wih.T (6 WMMA N-tiles, K=32); GRU gates
// elementwise in D-layout (r/z/n tiles align positionally); h_out transposed
// through LDS (hc rows of 34 = [h|ldI|ldR]); one WMMA tile for pred_I||pred_R
// (cols 0-7 = WI rows, 8-15 = WR rows; last_diff cols added as rank-2 fixup);
// per-lane Wsir + SIR recurrence.
// ---------------------------------------------------------------------------
__global__ __launch_bounds__(128) void k_head(const float* __restrict__ agg2,
    const float* __restrict__ X,
    const float* __restrict__ Wih, const float* __restrict__ bih,
    const float* __restrict__ bhh,
    const float* __restrict__ WI, const float* __restrict__ bIv,
    const float* __restrict__ WR, const float* __restrict__ bRv,
    const float* __restrict__ Wsir, const float* __restrict__ bsir,
    const float* __restrict__ states, const float* __restrict__ Nvec,
    float* __restrict__ out)
{
    __shared__ float hsh[4][16 * 34];                       // per-wave hc tiles
    const int lane = threadIdx.x & 31;
    const int wid  = threadIdx.x >> 5;
    const int tile = blockIdx.x * 4 + wid;
    const int base = tile * 16;
    const int mrow = lane & 15;
    const int kb   = (lane >> 4) * 2;
    const int mo   = (lane >> 4) * 8;

    v2f a[8];
    {
        const float* hp = agg2 + (size_t)(base + mrow) * 32 + kb;
#pragma unroll
        for (int ks = 0; ks < 8; ++ks) {
            a[ks].x = elu1(hp[ks * 4]);
            a[ks].y = elu1(hp[ks * 4 + 1]);
        }
    }
    v8f acc[6];
#pragma unroll
    for (int g = 0; g < 6; ++g) {
        const int col = g * 16 + mrow;
        const float* wr = Wih + (size_t)col * 32 + kb;
        v8f c = {};
#pragma unroll
        for (int ks = 0; ks < 8; ++ks) {
            v2f bv; bv.x = wr[ks * 4]; bv.y = wr[ks * 4 + 1];
            c = __builtin_amdgcn_wmma_f32_16x16x4_f32(false, a[ks], false, bv,
                                                      (short)0, c, false, false);
        }
        acc[g] = c;
    }
    // GRU gates (h_prev == 0, so gru_Whh drops out): tiles {0,1}=r {2,3}=z {4,5}=n
#pragma unroll
    for (int j = 0; j < 2; ++j) {
        const int c = j * 16 + mrow;
        const float bir = bih[c],      bhr = bhh[c];
        const float biz = bih[32 + c], bhz = bhh[32 + c];
        const float bin = bih[64 + c], bhn = bhh[64 + c];
#pragma unroll
        for (int v = 0; v < 8; ++v) {
            const float r  = sigm(acc[j][v] + bir + bhr);
            const float zg = sigm(acc[2 + j][v] + biz + bhz);
            const float nv = tanhf(acc[4 + j][v] + bin + r * bhn);
            hsh[wid][(mo + v) * 34 + c] = (1.f - zg) * nv;  // node-major transpose
        }
    }
    if (lane < 16) {                                        // last_diff_I/R columns
        const int nf = base + lane;
        const int bb = nf / NUM_NODES_C;
        const int nn = nf - bb * NUM_NODES_C;
        const float* xl = X + ((size_t)(bb * T_C + (T_C - 1)) * NUM_NODES_C + nn) * NF_C;
        hsh[wid][lane * 34 + 32] = xl[1];
        hsh[wid][lane * 34 + 33] = xl[2];
    }
    __syncthreads();

    // pred_I || pred_R in one WMMA tile (K = 32 part), full EXEC here.
    {
        v2f a2[8];
#pragma unroll
        for (int ks = 0; ks < 8; ++ks) {
            a2[ks].x = hsh[wid][mrow * 34 + ks * 4 + kb];
            a2[ks].y = hsh[wid][mrow * 34 + ks * 4 + kb + 1];
        }
        const int   r8 = mrow & 7;
        const int   wh = mrow >> 3;                         // 0 = I, 1 = R
        const float* Wp = ((mrow < 8) ? WI : WR) + (size_t)r8 * 34 + kb;
        v8f c = {};
#pragma unroll
        for (int ks = 0; ks < 8; ++ks) {
            v2f bv; bv.x = Wp[ks * 4]; bv.y = Wp[ks * 4 + 1];
            c = __builtin_amdgcn_wmma_f32_16x16x4_f32(false, a2[ks], false, bv,
                                                      (short)0, c, false, false);
        }
        const float bc  = (mrow < 8) ? bIv[r8] : bRv[r8];
        const float w32 = Wp[32 - kb], w33 = Wp[33 - kb];   // last_diff weights
#pragma unroll
        for (int v = 0; v < 8; ++v) {
            const int node = mo + v;
            const float ldI = hsh[wid][node * 34 + 32];
            const float ldR = hsh[wid][node * 34 + 33];
            out[(size_t)(base + node) * 16 + 2 * r8 + wh] =
                c[v] + bc + ldI * w32 + ldR * w33;
        }
    }

    // SIR recurrence tail: one lane per node.
    if (lane < 16) {
        const int nf = base + lane;
        float s0 = bsir[0], s1 = bsir[1];
#pragma unroll
        for (int cc = 0; cc < 34; ++cc) {
            const float h = hsh[wid][lane * 34 + cc];
            s0 += h * Wsir[cc];
            s1 += h * Wsir[34 + cc];
        }
        const float alpha = sigm(s0), beta = sigm(s1);
        float I = states[(size_t)nf * 2 + 0];
        float R = states[(size_t)nf * 2 + 1];
        const float Nv = Nvec[nf % NUM_NODES_C];
        float* po = out + (size_t)NTOT_C * 16 + (size_t)nf * 16;
#pragma unroll
        for (int h = 0; h < HOR_C; ++h) {
            const float S  = Nv - I - R;
            const float dI = alpha * I * (S / Nv) - beta * I;
            const float dR = beta * I;
            po[2 * h]     = dI;
            po[2 * h + 1] = dR;
            I += dI; R += dR;
        }
    }
}

// ---------------------------------------------------------------------------
extern "C" void kernel_launch(void* const* d_in, const int* in_sizes, int n_in,
                              void* d_out, int out_size, void* d_ws, size_t ws_size,
                              hipStream_t stream)
{
    const float* X      = (const float*)d_in[0];
    const int*   adj    = (const int*)d_in[1];
    const float* states = (const float*)d_in[2];
    const float* Nvec   = (const float*)d_in[3];
    const float* fc1_W  = (const float*)d_in[4];
    const float* fc1_b  = (const float*)d_in[5];
    const float* att1_W = (const float*)d_in[6];
    const float* att1_b = (const float*)d_in[7];
    const float* fc2_W  = (const float*)d_in[8];
    const float* fc2_b  = (const float*)d_in[9];
    const float* att2_W = (const float*)d_in[10];
    const float* att2_b = (const float*)d_in[11];
    const float* gWih   = (const float*)d_in[12];
    /* d_in[13] = gru_Whh: unused (h_prev == 0) */
    const float* gbih   = (const float*)d_in[14];
    const float* gbhh   = (const float*)d_in[15];
    const float* WI     = (const float*)d_in[16];
    const float* bI     = (const float*)d_in[17];
    const float* WR     = (const float*)d_in[18];
    const float* bR     = (const float*)d_in[19];
    const float* Wsir   = (const float*)d_in[20];
    const float* bsir   = (const float*)d_in[21];
    float* out = (float*)d_out;

    const int nE = in_sizes[1] / 2;
    const int* src = adj;
    const int* dst = adj + nE;

    // workspace: z (25.6MB) | agg1 (25.6MB) | agg2 (25.6MB) — L2-resident tables
    float* z    = (float*)d_ws;
    float* agg1 = z    + (size_t)NTOT_C * 32;
    float* agg2 = agg1 + (size_t)NTOT_C * 32;

    const dim3 gN(NTOT_C / 64), bN(128);                    // 12500 tiles, 4 waves/block
    const dim3 gE((nE + 7) / 8), bE(256);                   // 8 edges (waves) per block

    k_gemm1<<<gN, bN, 0, stream>>>(X, fc1_W, fc1_b, z, agg1);
    k_edge <<<gE, bE, 0, stream>>>(src, dst, z, att1_W, att1_b, agg1, nE);
    k_gemm2<<<gN, bN, 0, stream>>>(agg1, fc2_W, fc2_b, z, agg2);
    k_edge <<<gE, bE, 0, stream>>>(src, dst, z, att2_W, att2_b, agg2, nE);
    k_head <<<gN, bN, 0, stream>>>(agg2, X, gWih, gbih, gbhh,
                                   WI, bI, WR, bR, Wsir, bsir, states, Nvec, out);
}